// Sampling_27041114096411
// MI455X (gfx1250) — compile-verified
//
#include <hip/hip_runtime.h>
#include <hip/hip_bf16.h>
#include <stdint.h>

// ---------------- problem constants (match reference) ----------------
#define BATCH   32
#define NPTS    131072
#define KSAMP   2048
#define NWGB    16                    // workgroups cooperating per batch
#define SLICE   (NPTS / NWGB)         // 8192 points owned per WG
#define BLOCK   256                   // 8 wave32 waves
#define PPT     (SLICE / BLOCK)       // 32 points per thread -> coords + min_d in VGPRs
#define CHUNK   2048                  // staging bounce-buffer, 24 KB LDS
#define NCHUNK  (SLICE / CHUNK)       // 4 staging rounds
#define CPT     (CHUNK / BLOCK)       // 8 points per thread per round

// workspace layout
//   [0, 4096)        : u32 arrival counters, one per batch, stride 32 u32 (128B)
//   [4096, 4096+8KB) : u64 packed partials, [batch][2 (double buffer)][NWGB]

__global__ void fps_init_ws(unsigned* counters) {
    int i = blockIdx.x * blockDim.x + threadIdx.x;
    if (i < BATCH * 32) counters[i] = 0u;
}

__global__ __launch_bounds__(BLOCK)
void fps_kernel(const float* __restrict__ x,
                const int*   __restrict__ start_idx,
                int*         __restrict__ out,
                unsigned*    __restrict__ counters,
                unsigned long long* __restrict__ partials)
{
    __shared__ float sx[CHUNK], sy[CHUNK], sz[CHUNK];   // 24 KB staging buffer
    __shared__ unsigned long long s_wave[BLOCK / 32];
    __shared__ float s_cx, s_cy, s_cz;

    const int wg   = blockIdx.x % NWGB;        // workgroup-in-batch
    const int b    = blockIdx.x / NWGB;
    const int tid  = threadIdx.x;
    const int lane = tid & 31;                 // wave32
    const int wv   = tid >> 5;

    const float* xb = x + (size_t)b * NPTS * 3;
    const int sliceBase = wg * SLICE;

    // Register-resident point slice and min-dist state (~150 VGPRs/thread).
    float pxs[PPT], pys[PPT], pzs[PPT], mds[PPT];

    // ---- staging: async global->LDS DMA (ASYNCcnt) in 24 KB chunks, then
    //      LDS -> VGPRs. Each thread reads only the data its own asyncs wrote,
    //      so s_wait_asynccnt 0 alone orders the round trip. Point r (0..31)
    //      of this thread is slice index p = tid + r*BLOCK. ----
    for (int c = 0; c < NCHUNK; ++c) {
#pragma unroll
        for (int i = 0; i < CPT; ++i) {
            int q = tid + i * BLOCK;                         // index within chunk
            unsigned g12 = (unsigned)((sliceBase + c * CHUNK + q) * 12);
            unsigned lx = (unsigned)(size_t)(void*)&sx[q];
            unsigned ly = (unsigned)(size_t)(void*)&sy[q];
            unsigned lz = (unsigned)(size_t)(void*)&sz[q];
            asm volatile("global_load_async_to_lds_b32 %0, %1, %2"
                         :: "v"(lx), "v"(g12), "s"(xb) : "memory");
            asm volatile("global_load_async_to_lds_b32 %0, %1, %2 offset:4"
                         :: "v"(ly), "v"(g12), "s"(xb) : "memory");
            asm volatile("global_load_async_to_lds_b32 %0, %1, %2 offset:8"
                         :: "v"(lz), "v"(g12), "s"(xb) : "memory");
        }
        asm volatile("s_wait_asynccnt 0" ::: "memory");
#pragma unroll
        for (int i = 0; i < CPT; ++i) {
            int r = c * CPT + i;                             // 0..31
            int q = tid + i * BLOCK;
            pxs[r] = sx[q];
            pys[r] = sy[q];
            pzs[r] = sz[q];
        }
    }
#pragma unroll
    for (int r = 0; r < PPT; ++r) mds[r] = 3.402823466e38f;

    const int cur0 = start_idx[b];
    if (wg == 0 && tid == 0) out[(size_t)b * KSAMP] = cur0;  // idxs[0] = start_idx
    if (tid == 0) {
        const float* c = xb + (size_t)cur0 * 3;
        s_cx = c[0]; s_cy = c[1]; s_cz = c[2];
    }
    __syncthreads();

    unsigned*           ctr   = counters + b * 32;
    unsigned long long* bpart = partials + (size_t)b * 2 * NWGB;

    for (int k = 1; k < KSAMP; ++k) {
        const float cx = s_cx, cy = s_cy, cz = s_cz;

        // Pure-VALU steady state: squared distance (sqrt elided: monotone),
        // running min, argmax index as inline constant (0..31) via cndmask.
        // Ascending scan order + strict '>' keeps first-occurrence ties.
        float bestv = -1.0f;
        int   bestr = 0;
#pragma unroll
        for (int r = 0; r < PPT; ++r) {
            float dx = pxs[r] - cx;
            float dy = pys[r] - cy;
            float dz = pzs[r] - cz;
            float d2 = __builtin_fmaf(dx, dx, __builtin_fmaf(dy, dy, dz * dz));
            float m  = __builtin_fminf(mds[r], d2);
            mds[r] = m;
            if (m > bestv) { bestv = m; bestr = r; }
        }
        // pack once per thread: (min_d bits << 32) | ~global_idx  → u64 max == argmax
        unsigned long long best =
            ((unsigned long long)__float_as_uint(bestv) << 32) |
            (unsigned)~(unsigned)(sliceBase + tid + bestr * BLOCK);

        // wave32 shuffle reduction
#pragma unroll
        for (int off = 16; off > 0; off >>= 1) {
            unsigned long long o = __shfl_down(best, off, 32);
            best = o > best ? o : best;
        }
        if (lane == 0) s_wave[wv] = best;
        __syncthreads();

        if (tid == 0) {
            unsigned long long wbest = s_wave[0];
#pragma unroll
            for (int w = 1; w < BLOCK / 32; ++w)
                wbest = s_wave[w] > wbest ? s_wave[w] : wbest;

            // publish partial (double-buffered on k parity), arrive, spin, combine
            bpart[(k & 1) * NWGB + wg] = wbest;
            __hip_atomic_fetch_add(ctr, 1u, __ATOMIC_RELEASE, __HIP_MEMORY_SCOPE_AGENT);
            const unsigned target = (unsigned)(NWGB * k);
            while (__hip_atomic_load(ctr, __ATOMIC_ACQUIRE, __HIP_MEMORY_SCOPE_AGENT) < target)
                __builtin_amdgcn_s_sleep(1);

            const unsigned long long* pp = bpart + (k & 1) * NWGB;
            unsigned long long g = 0ull;
#pragma unroll
            for (int w = 0; w < NWGB; ++w) {
                unsigned long long v = pp[w];
                g = v > g ? v : g;
            }
            const int nidx = (int)~(unsigned)g;              // unpack winning index
            if (wg == 0) out[(size_t)b * KSAMP + k] = nidx;
            const float* c = xb + (size_t)nidx * 3;          // broadcast next centroid
            s_cx = c[0]; s_cy = c[1]; s_cz = c[2];
        }
        __syncthreads();                                     // single end-of-step barrier
    }
}

extern "C" void kernel_launch(void* const* d_in, const int* in_sizes, int n_in,
                              void* d_out, int out_size, void* d_ws, size_t ws_size,
                              hipStream_t stream)
{
    const float* x         = (const float*)d_in[0];
    const int*   start_idx = (const int*)d_in[1];
    int*         out       = (int*)d_out;

    unsigned*           counters = (unsigned*)d_ws;
    unsigned long long* partials = (unsigned long long*)((char*)d_ws + 4096);

    // zero per-batch barrier counters (stream-ordered before main kernel)
    fps_init_ws<<<(BATCH * 32 + 255) / 256, 256, 0, stream>>>(counters);

    fps_kernel<<<dim3(BATCH * NWGB), dim3(BLOCK), 0, stream>>>(
        x, start_idx, out, counters, partials);
}